// VectorQuantizer_35459249996606
// MI455X (gfx1250) — compile-verified
//
#include <hip/hip_runtime.h>

typedef float v2f __attribute__((ext_vector_type(2)));
typedef float v8f __attribute__((ext_vector_type(8)));

#define EMB_NUM 512
#define EMB_DIM 8
#define NVEC (65536 * 8)        // 524288 rows of dim 8
#define NELEM (65536 * 64)      // 4194304 output elements (z_q_st)
#define ROWS_PER_BLOCK 128      // 8 waves * 16 rows
#define NBLOCKS (NVEC / ROWS_PER_BLOCK)  // 4096
#define NTILES (EMB_NUM / 16)   // 32 code tiles
#define BETA 0.25f

// Main kernel: per-wave 16-row tile, WMMA f32 16x16x4 scores with
// software-pipelined LDS B-operand fetch, running argmax, gather winning
// code rows, per-block loss partial.
__global__ __launch_bounds__(256) void vq_wmma_kernel(
    const float* __restrict__ z, const float* __restrict__ emb,
    float* __restrict__ out, float* __restrict__ partials) {
  __shared__ float s_emb[EMB_NUM * EMB_DIM];  // 16 KB codebook
  __shared__ float s_hn[EMB_NUM];             // 0.5*||e_k||^2
  __shared__ float s_err[8];

  const int tid = threadIdx.x;

  // Cooperative codebook load (1024 float4 / 256 threads).
  {
    const float4* src = (const float4*)emb;
    float4* dst = (float4*)s_emb;
    #pragma unroll
    for (int j = 0; j < 4; ++j) dst[tid + j * 256] = src[tid + j * 256];
  }
  __syncthreads();
  // Half squared norms.
  #pragma unroll
  for (int kk = 0; kk < 2; ++kk) {
    const int k = tid + kk * 256;
    const float* e = &s_emb[k * EMB_DIM];
    float s = 0.f;
    #pragma unroll
    for (int d = 0; d < EMB_DIM; ++d) s += e[d] * e[d];
    s_hn[k] = 0.5f * s;
  }
  __syncthreads();

  const int wave = tid >> 5;
  const int lane = tid & 31;
  const int m = lane & 15;
  const bool hi = lane >= 16;
  const int rowbase = blockIdx.x * ROWS_PER_BLOCK + wave * 16;

  // A operands (16x4 f32 layout: VGPR0 = K0(lo)/K2(hi), VGPR1 = K1(lo)/K3(hi)).
  const float* zr = z + (size_t)(rowbase + m) * EMB_DIM;
  const int koff = hi ? 2 : 0;
  v2f a1, a2;
  a1[0] = zr[koff + 0]; a1[1] = zr[koff + 1];   // K 0..3
  a2[0] = zr[koff + 4]; a2[1] = zr[koff + 5];   // K 4..7

  // Partial ||z_row||^2: lo lane holds dims {0,1,4,5}, hi lane {2,3,6,7}.
  float p = a1[0]*a1[0] + a1[1]*a1[1] + a2[0]*a2[0] + a2[1]*a2[1];
  float sz = p + __shfl_xor(p, 16, 32);   // full ||z_{rowbase+m}||^2

  float best[8];
  int   bidx[8];
  #pragma unroll
  for (int v = 0; v < 8; ++v) { best[v] = -3.4e38f; bidx[v] = 0; }

  // Software-pipelined B/hn fetch: prologue loads tile 0.
  const float* eb = &s_emb[(size_t)m * EMB_DIM + koff];
  v2f b1, b2;
  b1[0] = eb[0]; b1[1] = eb[1];
  b2[0] = eb[4]; b2[1] = eb[5];
  float hn = s_hn[m];

  #pragma unroll 2
  for (int t = 0; t < NTILES; ++t) {
    // Issue next tile's LDS loads (wraps harmlessly on the last iteration).
    const int tn = (t + 1) & (NTILES - 1);
    const float* en = &s_emb[(size_t)(tn * 16 + m) * EMB_DIM + koff];
    v2f nb1, nb2;
    nb1[0] = en[0]; nb1[1] = en[1];
    nb2[0] = en[4]; nb2[1] = en[5];
    const float nhn = s_hn[tn * 16 + m];

    const int n = t * 16 + m;  // this lane's code column
    v8f c = {};
    c = __builtin_amdgcn_wmma_f32_16x16x4_f32(false, a1, false, b1,
                                              (short)0, c, false, false);
    c = __builtin_amdgcn_wmma_f32_16x16x4_f32(false, a2, false, b2,
                                              (short)0, c, false, false);
    #pragma unroll
    for (int v = 0; v < 8; ++v) {
      const float s = c[v] - hn;           // z.e - 0.5||e||^2
      const bool better = s > best[v];     // strict: earlier n wins ties
      bidx[v] = better ? n : bidx[v];
      best[v] = better ? s : best[v];
    }
    b1 = nb1; b2 = nb2; hn = nhn;
  }

  // Cross-lane argmax within each 16-lane half (same row per C slot).
  #pragma unroll
  for (int v = 0; v < 8; ++v) {
    #pragma unroll
    for (int d = 8; d >= 1; d >>= 1) {
      const float ob = __shfl_xor(best[v], d, 32);
      const int   oi = __shfl_xor(bidx[v], d, 32);
      const bool take = (ob > best[v]) || (ob == best[v] && oi < bidx[v]);
      best[v] = take ? ob : best[v];
      bidx[v] = take ? oi : bidx[v];
    }
  }

  const float szo = __shfl_xor(sz, 8, 32);  // ||z||^2 of row (m^8)
  float errc = 0.f;
  if (m < 8) {
    // Writer lane m handles C slot v=m: row rowbase+m (lo) / rowbase+m+8 (hi).
    int idxf = bidx[0]; float bf = best[0];
    #pragma unroll
    for (int v = 1; v < 8; ++v) { if (m == v) { idxf = bidx[v]; bf = best[v]; } }
    const int r = rowbase + m + (hi ? 8 : 0);
    const float sz_row = hi ? szo : sz;
    errc = sz_row - 2.f * bf;              // ||z - e_idx||^2
    const float4* e4 = (const float4*)&s_emb[(size_t)idxf * EMB_DIM];
    float4* o4 = (float4*)(out + (size_t)r * EMB_DIM);
    o4[0] = e4[0];
    o4[1] = e4[1];
  }
  // Wave-reduce loss contribution.
  #pragma unroll
  for (int d = 16; d >= 1; d >>= 1) errc += __shfl_xor(errc, d, 32);
  if (lane == 0) s_err[wave] = errc;
  __syncthreads();
  if (tid == 0) {
    float s = 0.f;
    #pragma unroll
    for (int w = 0; w < 8; ++w) s += s_err[w];
    partials[blockIdx.x] = s;
  }
}

// Deterministic final reduction of the 4096 block partials -> scalar loss.
__global__ __launch_bounds__(256) void vq_loss_kernel(
    const float* __restrict__ partials, float* __restrict__ loss_out) {
  __shared__ float s[256];
  float acc = 0.f;
  for (int i = threadIdx.x; i < NBLOCKS; i += 256) acc += partials[i];
  s[threadIdx.x] = acc;
  __syncthreads();
  for (int stride = 128; stride >= 1; stride >>= 1) {
    if (threadIdx.x < stride) s[threadIdx.x] += s[threadIdx.x + stride];
    __syncthreads();
  }
  if (threadIdx.x == 0)
    loss_out[0] = s[0] * ((1.0f + BETA) / (float)NELEM);
}

extern "C" void kernel_launch(void* const* d_in, const int* in_sizes, int n_in,
                              void* d_out, int out_size, void* d_ws, size_t ws_size,
                              hipStream_t stream) {
  const float* z   = (const float*)d_in[0];   // [65536, 64] f32
  const float* emb = (const float*)d_in[1];   // [512, 8]  f32
  float* out = (float*)d_out;                 // z_q_st (4194304) ++ loss (1)
  float* partials = (float*)d_ws;             // 4096 floats of scratch

  vq_wmma_kernel<<<NBLOCKS, 256, 0, stream>>>(z, emb, out, partials);
  vq_loss_kernel<<<1, 256, 0, stream>>>(partials, out + NELEM);
}